// SimAwareHANLayer_84138409328873
// MI455X (gfx1250) — compile-verified
//
#include <hip/hip_runtime.h>
#include <math.h>

#define NEG_SLOPE 0.2f

typedef float v2f __attribute__((ext_vector_type(2)));
typedef float v8f __attribute__((ext_vector_type(8)));

// ---------------------------------------------------------------------------
// Generic fill
// ---------------------------------------------------------------------------
__global__ void fill_f32(float* __restrict__ p, float v, int n) {
  int i = blockIdx.x * blockDim.x + threadIdx.x;
  if (i < n) p[i] = v;
}

// ---------------------------------------------------------------------------
// Y(N,128) = X(N,128) @ W(128,128), exact fp32 via V_WMMA_F32_16X16X4_F32.
// Block = 256 threads = 8 waves. Block covers 16 rows; wave w covers cols
// [16w,16w+16). Each wave: 32 k-steps of 16x16x4 fp32 WMMA.
// A layout (16x4 f32): lanes 0-15 = M, VGPR0/1 = K{0,1}; lanes 16-31 = K{2,3}.
// B layout (4x16 f32): N striped across lanes, K across VGPRs/lane halves.
// C/D: VGPR r -> M = r (lanes 0-15) / r+8 (lanes 16-31), N = lane%16.
// ---------------------------------------------------------------------------
__global__ void gemm_xw_wmma(const float* __restrict__ X, const float* __restrict__ W,
                             float* __restrict__ Y, int N) {
  const int wave   = threadIdx.x >> 5;
  const int lane   = threadIdx.x & 31;
  const int laneLo = lane & 15;
  const int laneHi = lane >> 4;
  const int row0   = blockIdx.x << 4;
  const int col0   = wave << 4;

  int arow = row0 + laneLo;
  if (arow >= N) arow = N - 1;               // clamped loads; tail rows never stored
  const float* __restrict__ xr = X + (size_t)arow * 128;
  const float* __restrict__ wc = W + col0 + laneLo;

  // gfx1250 global_prefetch_b8 hints for the A row and B column tile
  __builtin_prefetch(xr, 0, 0);
  __builtin_prefetch(wc, 0, 0);

  v8f acc = {0.f, 0.f, 0.f, 0.f, 0.f, 0.f, 0.f, 0.f};
#pragma unroll 8
  for (int k = 0; k < 128; k += 4) {
    const int ka = k + (laneHi << 1);
    v2f a, b;
    a.x = xr[ka];
    a.y = xr[ka + 1];
    b.x = wc[(size_t)ka * 128];
    b.y = wc[(size_t)(ka + 1) * 128];
    acc = __builtin_amdgcn_wmma_f32_16x16x4_f32(false, a, false, b,
                                                (short)0, acc, false, false);
  }

  float* __restrict__ ybase = Y + (size_t)(row0 + (laneHi << 3)) * 128 + col0 + laneLo;
  if (row0 + 16 <= N) {
    // full tile: branchless, clusterable stores
#pragma unroll
    for (int r = 0; r < 8; ++r) ybase[(size_t)r * 128] = acc[r];
  } else {
#pragma unroll
    for (int r = 0; r < 8; ++r) {
      const int row = row0 + r + (laneHi << 3);
      if (row < N) Y[(size_t)row * 128 + col0 + laneLo] = acc[r];
    }
  }
}

// ---------------------------------------------------------------------------
// e[n,h] = sum_d F[n,h,d] * a[h,d]      (H=8, D=16, hidden=128)
// one thread per (n,h); F + idx*16 == F[n*128 + h*16]
// ---------------------------------------------------------------------------
__global__ void attn_dots(const float* __restrict__ F, const float* __restrict__ avec,
                          float* __restrict__ e, int count /* = N*8 */) {
  int idx = blockIdx.x * blockDim.x + threadIdx.x;
  if (idx >= count) return;
  const int h = idx & 7;
  const float4* __restrict__ fr = (const float4*)(F + (size_t)idx * 16);
  const float4* __restrict__ ar = (const float4*)(avec + h * 16);
  float s = 0.f;
#pragma unroll
  for (int q = 0; q < 4; ++q) {
    float4 f = fr[q], a = ar[q];
    s += f.x * a.x + f.y * a.y + f.z * a.z + f.w * a.w;
  }
  e[idx] = s;
}

// Exact float atomic max via sign-split int/uint atomics (NaN-free inputs).
__device__ inline void atomicMaxF32(float* addr, float val) {
  if (val >= 0.f)
    atomicMax((int*)addr, __float_as_int(val));
  else
    atomicMin((unsigned int*)addr, (unsigned int)__float_as_int(val));
}

// ---------------------------------------------------------------------------
// Pass 1: e = leaky_relu(el[src]+er[dst]); segment max over dst.
// ---------------------------------------------------------------------------
__global__ void edge_max(const int* __restrict__ src, const int* __restrict__ dst,
                         const float* __restrict__ el, const float* __restrict__ er,
                         float* __restrict__ emax, int E) {
  int idx = blockIdx.x * blockDim.x + threadIdx.x;
  if (idx >= E * 8) return;
  const int ed = idx >> 3, h = idx & 7;
  const int si = src[ed], di = dst[ed];
  float e = el[si * 8 + h] + er[di * 8 + h];
  e = e > 0.f ? e : NEG_SLOPE * e;
  atomicMaxF32(&emax[di * 8 + h], e);
}

// ---------------------------------------------------------------------------
// Pass 2: ez = exp(e - emax[dst]); segment sum into denom; stash ez per edge.
// ---------------------------------------------------------------------------
__global__ void edge_expsum(const int* __restrict__ src, const int* __restrict__ dst,
                            const float* __restrict__ el, const float* __restrict__ er,
                            const float* __restrict__ emax, float* __restrict__ denom,
                            float* __restrict__ ez, int E) {
  int idx = blockIdx.x * blockDim.x + threadIdx.x;
  if (idx >= E * 8) return;
  const int ed = idx >> 3, h = idx & 7;
  const int si = src[ed], di = dst[ed];
  float e = el[si * 8 + h] + er[di * 8 + h];
  e = e > 0.f ? e : NEG_SLOPE * e;
  const float z = expf(e - emax[di * 8 + h]);
  ez[idx] = z;
  atomicAdd(&denom[di * 8 + h], z);
}

// ---------------------------------------------------------------------------
// Pass 3: out[dst] += (ez/denom[dst]) * fs[src].  One wave per edge,
// float4 per lane -> coalesced 512B row gather, 4 float atomics per lane.
// ---------------------------------------------------------------------------
__global__ void edge_aggregate(const int* __restrict__ src, const int* __restrict__ dst,
                               const float* __restrict__ ez, const float* __restrict__ denom,
                               const float* __restrict__ fs, float* __restrict__ out, int E) {
  const int lane = threadIdx.x & 31;
  const int ed = blockIdx.x * 8 + (threadIdx.x >> 5);
  if (ed >= E) return;
  const int si = src[ed], di = dst[ed];
  const int h = lane >> 2;                       // 4 channels per lane, 16 per head
  const float alpha = ez[(size_t)ed * 8 + h] / denom[di * 8 + h];
  const float4 f = ((const float4*)(fs + (size_t)si * 128))[lane];
  float* o = out + (size_t)di * 128 + (lane << 2);
  atomicAdd(o + 0, f.x * alpha);
  atomicAdd(o + 1, f.y * alpha);
  atomicAdd(o + 2, f.z * alpha);
  atomicAdd(o + 3, f.w * alpha);
}

// ---------------------------------------------------------------------------
// Semantic attention score: s[n*R + slot] = tanh((o[n]+b_rel) @ W1 + b1) @ W2 + b2
// One block (128 threads) per node; feature row staged in LDS.
// ---------------------------------------------------------------------------
__global__ void fuse_score(const float* __restrict__ o, const float* __restrict__ brel,
                           const float* __restrict__ W1, const float* __restrict__ b1,
                           const float* __restrict__ W2, const float* __restrict__ b2,
                           float* __restrict__ s, int R, int slot) {
  __shared__ float f[128];
  __shared__ float partial[128];
  const int n = blockIdx.x;
  const int c = threadIdx.x;
  f[c] = o[(size_t)n * 128 + c] + brel[c];
  __syncthreads();
  float acc = b1[c];
#pragma unroll 8
  for (int k = 0; k < 128; ++k) acc += f[k] * W1[k * 128 + c];
  partial[c] = tanhf(acc) * W2[c];
  __syncthreads();
  for (int off = 64; off > 0; off >>= 1) {
    if (c < off) partial[c] += partial[c + off];
    __syncthreads();
  }
  if (c == 0) s[(size_t)n * R + slot] = partial[0] + b2[0];
}

// ---------------------------------------------------------------------------
// Softmax over relations + weighted sum:  out[n,c] = sum_r w[n,r]*(o_r[n,c]+b_r[c])
// ---------------------------------------------------------------------------
__global__ void fuse_combine(const float* __restrict__ o0, const float* __restrict__ o1,
                             const float* o2,
                             const float* __restrict__ bb0, const float* __restrict__ bb1,
                             const float* bb2,
                             const float* __restrict__ s, float* __restrict__ out,
                             int N, int R) {
  int idx = blockIdx.x * blockDim.x + threadIdx.x;
  if (idx >= N * 128) return;
  const int n = idx >> 7, c = idx & 127;
  const float s0 = s[(size_t)n * R + 0];
  const float s1 = s[(size_t)n * R + 1];
  const float s2 = (R > 2) ? s[(size_t)n * R + 2] : -INFINITY;
  const float m = fmaxf(s0, fmaxf(s1, s2));
  const float w0 = expf(s0 - m), w1 = expf(s1 - m);
  const float w2 = (R > 2) ? expf(s2 - m) : 0.f;
  const float inv = 1.f / (w0 + w1 + w2);
  float v = (o0[idx] + bb0[c]) * w0 + (o1[idx] + bb1[c]) * w1;
  if (R > 2) v += (o2[idx] + bb2[c]) * w2;
  out[idx] = v * inv;
}

// ---------------------------------------------------------------------------
// Launcher
// ---------------------------------------------------------------------------
extern "C" void kernel_launch(void* const* d_in, const int* in_sizes, int n_in,
                              void* d_out, int out_size, void* d_ws, size_t ws_size,
                              hipStream_t stream) {
  (void)n_in; (void)out_size; (void)ws_size;
  const float* h_drug = (const float*)d_in[0];
  const float* h_prot = (const float*)d_in[1];
  const float* h_side = (const float*)d_in[2];
  const float* W_rel  = (const float*)d_in[3];   // (5,128,128)
  const float* al_rel = (const float*)d_in[4];   // (5,8,16)
  const float* ar_rel = (const float*)d_in[5];   // (5,8,16)
  const float* b_rel  = (const float*)d_in[6];   // (5,128)
  const float* W1     = (const float*)d_in[7];   // (128,128)
  const float* b1     = (const float*)d_in[8];   // (128)
  const float* W2     = (const float*)d_in[9];   // (128,1)
  const float* b2     = (const float*)d_in[10];  // (1)

  const int* src[5] = {(const int*)d_in[11], (const int*)d_in[13], (const int*)d_in[15],
                       (const int*)d_in[17], (const int*)d_in[19]};
  const int* dst[5] = {(const int*)d_in[12], (const int*)d_in[14], (const int*)d_in[16],
                       (const int*)d_in[18], (const int*)d_in[20]};
  int Ed[5] = {in_sizes[11], in_sizes[13], in_sizes[15], in_sizes[17], in_sizes[19]};

  const int Nd = in_sizes[0] / 128;
  const int Np = in_sizes[1] / 128;
  const int Ns = in_sizes[2] / 128;

  // relation tables: 0=dd 1=dp 2=pd 3=pp 4=sd
  const float* hs_r[5] = {h_drug, h_drug, h_prot, h_prot, h_side};
  const int    Nsr[5]  = {Nd, Nd, Np, Np, Ns};
  const float* hd_r[5] = {h_drug, h_prot, h_drug, h_prot, h_drug};
  const int    Ndr[5]  = {Nd, Np, Nd, Np, Nd};

  // workspace layout (buffers reused across sequentially-processed relations)
  float* ws = (float*)d_ws;
  size_t off = 0;
  auto alloc = [&](size_t n) { float* p = ws + off; off += n; return p; };
  const int maxN = (Nd > Np) ? ((Nd > Ns) ? Nd : Ns) : ((Np > Ns) ? Np : Ns);
  int maxE = Ed[0];
  for (int r = 1; r < 5; ++r) if (Ed[r] > maxE) maxE = Ed[r];

  float* bufA  = alloc((size_t)maxN * 128);   // fs transform
  float* bufB  = alloc((size_t)maxN * 128);   // fd transform
  float* el    = alloc((size_t)maxN * 8);
  float* er    = alloc((size_t)maxN * 8);
  float* emax  = alloc((size_t)maxN * 8);
  float* denom = alloc((size_t)maxN * 8);
  float* ez    = alloc((size_t)maxE * 8);
  float* o_r[5];
  for (int r = 0; r < 5; ++r) o_r[r] = alloc((size_t)Ndr[r] * 128);
  float* s_drug = alloc((size_t)Nd * 3);
  float* s_prot = alloc((size_t)Np * 2);

  for (int r = 0; r < 5; ++r) {
    const float* Wr = W_rel + (size_t)r * 128 * 128;
    const bool same = (hs_r[r] == hd_r[r]);

    gemm_xw_wmma<<<(Nsr[r] + 15) / 16, 256, 0, stream>>>(hs_r[r], Wr, bufA, Nsr[r]);
    const float* fd = bufA;
    if (!same) {
      gemm_xw_wmma<<<(Ndr[r] + 15) / 16, 256, 0, stream>>>(hd_r[r], Wr, bufB, Ndr[r]);
      fd = bufB;
    }

    const int nl = Nsr[r] * 8;
    const int nr = Ndr[r] * 8;
    attn_dots<<<(nl + 255) / 256, 256, 0, stream>>>(bufA, al_rel + r * 128, el, nl);
    attn_dots<<<(nr + 255) / 256, 256, 0, stream>>>(fd,   ar_rel + r * 128, er, nr);

    fill_f32<<<(nr + 255) / 256, 256, 0, stream>>>(emax, -INFINITY, nr);
    fill_f32<<<(nr + 255) / 256, 256, 0, stream>>>(denom, 0.f, nr);
    const int no = Ndr[r] * 128;
    fill_f32<<<(no + 255) / 256, 256, 0, stream>>>(o_r[r], 0.f, no);

    const int ne = Ed[r] * 8;
    edge_max<<<(ne + 255) / 256, 256, 0, stream>>>(src[r], dst[r], el, er, emax, Ed[r]);
    edge_expsum<<<(ne + 255) / 256, 256, 0, stream>>>(src[r], dst[r], el, er, emax, denom, ez, Ed[r]);
    edge_aggregate<<<(Ed[r] + 7) / 8, 256, 0, stream>>>(src[r], dst[r], ez, denom, bufA, o_r[r], Ed[r]);
  }

  // semantic fusion: drug = stack(o_dd, o_pd, o_sd); protein = stack(o_dp, o_pp)
  fuse_score<<<Nd, 128, 0, stream>>>(o_r[0], b_rel + 0 * 128, W1, b1, W2, b2, s_drug, 3, 0);
  fuse_score<<<Nd, 128, 0, stream>>>(o_r[2], b_rel + 2 * 128, W1, b1, W2, b2, s_drug, 3, 1);
  fuse_score<<<Nd, 128, 0, stream>>>(o_r[4], b_rel + 4 * 128, W1, b1, W2, b2, s_drug, 3, 2);
  fuse_score<<<Np, 128, 0, stream>>>(o_r[1], b_rel + 1 * 128, W1, b1, W2, b2, s_prot, 2, 0);
  fuse_score<<<Np, 128, 0, stream>>>(o_r[3], b_rel + 3 * 128, W1, b1, W2, b2, s_prot, 2, 1);

  float* out_drug = (float*)d_out;
  float* out_prot = out_drug + (size_t)Nd * 128;
  fuse_combine<<<(Nd * 128 + 255) / 256, 256, 0, stream>>>(
      o_r[0], o_r[2], o_r[4], b_rel, b_rel + 2 * 128, b_rel + 4 * 128,
      s_drug, out_drug, Nd, 3);
  fuse_combine<<<(Np * 128 + 255) / 256, 256, 0, stream>>>(
      o_r[1], o_r[3], nullptr, b_rel + 1 * 128, b_rel + 3 * 128, nullptr,
      s_prot, out_prot, Np, 2);
}